// GoalPrediction_73967926771941
// MI455X (gfx1250) — compile-verified
//
#include <hip/hip_runtime.h>
#include <hip/hip_bf16.h>
#include <math.h>

// ---------------------------------------------------------------------------
// GoalPrediction pipeline for MI455X (gfx1250, wave32, WMMA).
// fp32 matrix path: V_WMMA_F32_16X16X4_F32. Deconv processes 2 batches per
// wave (8 WMMA chains sharing each weight tile: 6 b64 loads / 8 WMMAs) with
// software-pipelined operand loads. Activations channel-last (b, spatial, C);
// weights packed (tap, O, I); GroupNorm applied once per tensor outside the
// GEMMs; zero-padding taps read a zeroed scratch row.
// ---------------------------------------------------------------------------

typedef __attribute__((ext_vector_type(2))) float v2f;
typedef __attribute__((ext_vector_type(8))) float v8f;

#define WMMA_F32(A, B, C) \
    __builtin_amdgcn_wmma_f32_16x16x4_f32(false, (A), false, (B), (short)0, (C), false, false)

// --------------------------- layout-packing kernels --------------------------
// feat (B,256,16) -> ft (B,16,256)
__global__ void __launch_bounds__(256)
transpose_feat_kernel(const float* __restrict__ feat, float* __restrict__ ft) {
    const int idx = blockIdx.x * 256 + threadIdx.x;     // over B*16*256
    const int b = idx >> 12;
    const int p = (idx >> 8) & 15;
    const int c = idx & 255;
    ft[idx] = feat[((size_t)b * 256 + c) * 16 + p];
}

// fc_w (256,512) -> fcwt (512,256)
__global__ void __launch_bounds__(256)
transpose_fcw_kernel(const float* __restrict__ fcw, float* __restrict__ fcwt) {
    const int idx = blockIdx.x * 256 + threadIdx.x;     // over 512*256
    const int d = idx >> 8;
    const int c = idx & 255;
    fcwt[idx] = fcw[(size_t)c * 512 + d];
}

// dw (O,I,3,3) -> wp (9,O,I)
__global__ void __launch_bounds__(256)
pack_dw_kernel(const float* __restrict__ dw, float* __restrict__ wp, int O, int I) {
    const int idx  = blockIdx.x * 256 + threadIdx.x;    // over 9*O*I
    const int OI   = O * I;
    const int tap  = idx / OI;
    const int rem  = idx - tap * OI;
    wp[idx] = dw[(size_t)rem * 9 + tap];
}

__global__ void __launch_bounds__(256)
zero_fill_kernel(float* __restrict__ p, int n) {
    const int idx = blockIdx.x * 256 + threadIdx.x;
    if (idx < n) p[idx] = 0.0f;
}

// ------------------------- kernel 1: fc (1x1) + ReLU -------------------------
// x0t[b, n, d] = relu(sum_c ft[b,n,c] * fcwt[d,c] + fcb[d]); n = 16 positions.
__global__ void __launch_bounds__(32)
fc_relu_kernel(const float* __restrict__ ft, const float* __restrict__ fcwt,
               const float* __restrict__ fcb, float* __restrict__ x0t) {
    const int lane = threadIdx.x;
    const int m  = lane & 15;
    const int kk = (lane >> 4) << 1;
    const int mb = (lane >> 4) << 3;
    const int dg = blockIdx.x << 6;          // 64-channel output group
    const int b  = blockIdx.y;

    const float* br = ft + (size_t)b * 16 * 256 + m * 256 + kk;
    const float* w0 = fcwt + (size_t)(dg + 0 * 16 + m) * 256 + kk;
    const float* w1 = fcwt + (size_t)(dg + 1 * 16 + m) * 256 + kk;
    const float* w2 = fcwt + (size_t)(dg + 2 * 16 + m) * 256 + kk;
    const float* w3 = fcwt + (size_t)(dg + 3 * 16 + m) * 256 + kk;

    v8f acc0 = {}, acc1 = {}, acc2 = {}, acc3 = {};
    v2f Bc  = *(const v2f*)(br);
    v2f A0c = *(const v2f*)(w0);
    v2f A1c = *(const v2f*)(w1);
    v2f A2c = *(const v2f*)(w2);
    v2f A3c = *(const v2f*)(w3);
    for (int k0 = 4; k0 < 256; k0 += 4) {
        const v2f Bn  = *(const v2f*)(br + k0);
        const v2f A0n = *(const v2f*)(w0 + k0);
        const v2f A1n = *(const v2f*)(w1 + k0);
        const v2f A2n = *(const v2f*)(w2 + k0);
        const v2f A3n = *(const v2f*)(w3 + k0);
        acc0 = WMMA_F32(A0c, Bc, acc0);
        acc1 = WMMA_F32(A1c, Bc, acc1);
        acc2 = WMMA_F32(A2c, Bc, acc2);
        acc3 = WMMA_F32(A3c, Bc, acc3);
        Bc = Bn; A0c = A0n; A1c = A1n; A2c = A2n; A3c = A3n;
    }
    acc0 = WMMA_F32(A0c, Bc, acc0);
    acc1 = WMMA_F32(A1c, Bc, acc1);
    acc2 = WMMA_F32(A2c, Bc, acc2);
    acc3 = WMMA_F32(A3c, Bc, acc3);

    float* orow = x0t + ((size_t)b * 16 + m) * 512;
    #pragma unroll
    for (int j = 0; j < 8; ++j) {
        int d;
        d = dg + 0 * 16 + mb + j; orow[d] = fmaxf(acc0[j] + fcb[d], 0.0f);
        d = dg + 1 * 16 + mb + j; orow[d] = fmaxf(acc1[j] + fcb[d], 0.0f);
        d = dg + 2 * 16 + mb + j; orow[d] = fmaxf(acc2[j] + fcb[d], 0.0f);
        d = dg + 3 * 16 + mb + j; orow[d] = fmaxf(acc3[j] + fcb[d], 0.0f);
    }
}

// ------------------ kernel 2: per-(b,c) mean/rstd, channel-last --------------
__global__ void __launch_bounds__(256)
gn_stats_kernel(const float* __restrict__ x, float* __restrict__ mu,
                float* __restrict__ rstd, int C, int S) {
    const int gid = blockIdx.x * 256 + threadIdx.x;     // over B*C
    const int b = gid / C;
    const int c = gid - b * C;
    const float* xb = x + (size_t)b * S * C + c;
    float s = 0.0f, ss = 0.0f;
    for (int sp = 0; sp < S; ++sp) {
        const float v = xb[(size_t)sp * C];
        s += v; ss += v * v;
    }
    const float inv  = 1.0f / (float)S;
    const float mean = s * inv;
    const float var  = ss * inv - mean * mean;
    mu[gid]   = mean;
    rstd[gid] = rsqrtf(var + 1e-5f);
}

// --------------- kernel 3: apply GroupNorm affine in place ------------------
__global__ void __launch_bounds__(256)
gn_apply_kernel(float* __restrict__ x, const float* __restrict__ mu,
                const float* __restrict__ rstd, const float* __restrict__ gw,
                const float* __restrict__ gb, int C, int SC) {
    const size_t idx = (size_t)blockIdx.x * 256 + threadIdx.x;  // over B*S*C
    const int b = (int)(idx / SC);
    const int c = (int)(idx & (size_t)(C - 1));                 // C is pow2
    const int bc = b * C + c;
    x[idx] = (x[idx] - mu[bc]) * rstd[bc] * gw[c] + gb[c];
}

// --------------- kernel 4: deconv (stride-2, 3x3) + bias + ReLU --------------
// Gather form over parity classes; 2 batches per wave share each weight tile:
// per K-step 4 A-loads + 2 B-loads feed 8 WMMA chains. Out-of-range taps read
// a zero-filled scratch row.
__global__ void __launch_bounds__(32)
deconv_stage_kernel(const float* __restrict__ x, const float* __restrict__ wp,
                    const float* __restrict__ db, const float* __restrict__ zrow,
                    float* __restrict__ y, int Cin, int Cout, int Hin) {
    const int Hout = Hin * 2;
    const int nt   = (Hin * Hin) >> 4;   // spatial tiles per parity class
    const int lane = threadIdx.x;
    const int m  = lane & 15;
    const int kk = (lane >> 4) << 1;
    const int mb = (lane >> 4) << 3;
    const int b0 = blockIdx.y * 2;       // batch pair
    const int b1 = b0 + 1;

    const int tid = blockIdx.x;
    const int cg  = tid / (4 * nt);
    const int rem = tid - cg * 4 * nt;
    const int par = rem / nt;
    const int st  = rem - par * nt;
    const int pp  = par >> 1, qq = par & 1;
    const int o0  = cg << 6;

    const int t = st * 16 + m;           // lane's spatial index in parity class
    const int u = t / Hin, v = t - u * Hin;
    const int p = 2 * u + pp, q = 2 * v + qq;

    const float* xb0 = x + (size_t)b0 * Hin * Hin * Cin;
    const float* xb1 = x + (size_t)b1 * Hin * Hin * Cin;

    v8f acc0a = {}, acc1a = {}, acc2a = {}, acc3a = {};
    v8f acc0b = {}, acc1b = {}, acc2b = {}, acc3b = {};
    const int nh = pp ? 2 : 1;
    const int nw = qq ? 2 : 1;
    for (int th = 0; th < nh; ++th) {
        const int r  = pp ? (th ? 2 : 0) : 1;
        const int ih = pp ? (u + th)     : u;
        for (int tw = 0; tw < nw; ++tw) {
            const int s  = qq ? (tw ? 2 : 0) : 1;
            const int iw = qq ? (v + tw)     : v;
            const bool valid = (ih < Hin) && (iw < Hin);
            const size_t soff = (size_t)(valid ? (ih * Hin + iw) : 0) * Cin;
            const float* bra = (valid ? (xb0 + soff) : zrow) + kk;
            const float* brb = (valid ? (xb1 + soff) : zrow) + kk;
            const float* wtap = wp + (size_t)(r * 3 + s) * Cout * Cin;
            const float* w0 = wtap + (size_t)(o0 + 0 * 16 + m) * Cin + kk;
            const float* w1 = wtap + (size_t)(o0 + 1 * 16 + m) * Cin + kk;
            const float* w2 = wtap + (size_t)(o0 + 2 * 16 + m) * Cin + kk;
            const float* w3 = wtap + (size_t)(o0 + 3 * 16 + m) * Cin + kk;

            // software-pipelined K loop
            v2f Bca = *(const v2f*)(bra);
            v2f Bcb = *(const v2f*)(brb);
            v2f A0c = *(const v2f*)(w0);
            v2f A1c = *(const v2f*)(w1);
            v2f A2c = *(const v2f*)(w2);
            v2f A3c = *(const v2f*)(w3);
            for (int k0 = 4; k0 < Cin; k0 += 4) {
                const v2f Bna = *(const v2f*)(bra + k0);
                const v2f Bnb = *(const v2f*)(brb + k0);
                const v2f A0n = *(const v2f*)(w0 + k0);
                const v2f A1n = *(const v2f*)(w1 + k0);
                const v2f A2n = *(const v2f*)(w2 + k0);
                const v2f A3n = *(const v2f*)(w3 + k0);
                acc0a = WMMA_F32(A0c, Bca, acc0a);
                acc1a = WMMA_F32(A1c, Bca, acc1a);
                acc2a = WMMA_F32(A2c, Bca, acc2a);
                acc3a = WMMA_F32(A3c, Bca, acc3a);
                acc0b = WMMA_F32(A0c, Bcb, acc0b);
                acc1b = WMMA_F32(A1c, Bcb, acc1b);
                acc2b = WMMA_F32(A2c, Bcb, acc2b);
                acc3b = WMMA_F32(A3c, Bcb, acc3b);
                Bca = Bna; Bcb = Bnb;
                A0c = A0n; A1c = A1n; A2c = A2n; A3c = A3n;
            }
            acc0a = WMMA_F32(A0c, Bca, acc0a);
            acc1a = WMMA_F32(A1c, Bca, acc1a);
            acc2a = WMMA_F32(A2c, Bca, acc2a);
            acc3a = WMMA_F32(A3c, Bca, acc3a);
            acc0b = WMMA_F32(A0c, Bcb, acc0b);
            acc1b = WMMA_F32(A1c, Bcb, acc1b);
            acc2b = WMMA_F32(A2c, Bcb, acc2b);
            acc3b = WMMA_F32(A3c, Bcb, acc3b);
        }
    }
    const size_t spoff = ((size_t)p * Hout + q) * Cout;
    float* yra = y + (size_t)b0 * Hout * Hout * Cout + spoff;
    float* yrb = y + (size_t)b1 * Hout * Hout * Cout + spoff;
    #pragma unroll
    for (int j = 0; j < 8; ++j) {
        int o;
        o = o0 + 0 * 16 + mb + j;
        yra[o] = fmaxf(acc0a[j] + db[o], 0.0f);
        yrb[o] = fmaxf(acc0b[j] + db[o], 0.0f);
        o = o0 + 1 * 16 + mb + j;
        yra[o] = fmaxf(acc1a[j] + db[o], 0.0f);
        yrb[o] = fmaxf(acc1b[j] + db[o], 0.0f);
        o = o0 + 2 * 16 + mb + j;
        yra[o] = fmaxf(acc2a[j] + db[o], 0.0f);
        yrb[o] = fmaxf(acc2b[j] + db[o], 0.0f);
        o = o0 + 3 * 16 + mb + j;
        yra[o] = fmaxf(acc3a[j] + db[o], 0.0f);
        yrb[o] = fmaxf(acc3b[j] + db[o], 0.0f);
    }
}

// ------------- kernel 5: routed head + spatial softmax + coords --------------
__device__ __forceinline__ float block_reduce(float v, float* red, int tid, bool ismax) {
    red[tid] = v;
    __syncthreads();
    #pragma unroll
    for (int off = 128; off > 0; off >>= 1) {
        if (tid < off)
            red[tid] = ismax ? fmaxf(red[tid], red[tid + off]) : (red[tid] + red[tid + off]);
        __syncthreads();
    }
    const float r = red[0];
    __syncthreads();
    return r;
}

__global__ void __launch_bounds__(256)
head_softmax_kernel(const float* __restrict__ x3, const float* __restrict__ mu,
                    const float* __restrict__ rstd, const int* __restrict__ action,
                    const float* __restrict__ hgw, const float* __restrict__ hgb,
                    const float* __restrict__ hcw, const float* __restrict__ hcb,
                    float* __restrict__ out) {
    __shared__ float weff[5 * 64];
    __shared__ float beff[5];
    __shared__ float mus[64];
    __shared__ float rss[64];
    __shared__ float red[256];
    const int b   = blockIdx.x;
    const int tid = threadIdx.x;
    const int a   = action[b];

    for (int i = tid; i < 5 * 64; i += 256) {
        const int s = i >> 6, c = i & 63;
        weff[i] = hcw[(a * 5 + s) * 64 + c] * hgw[a * 64 + c];
    }
    if (tid < 64) {
        mus[tid] = mu[b * 64 + tid];
        rss[tid] = rstd[b * 64 + tid];
    }
    if (tid < 5) {
        float acc = hcb[a * 5 + tid];
        for (int c = 0; c < 64; ++c)
            acc += hcw[(a * 5 + tid) * 64 + c] * hgb[a * 64 + c];
        beff[tid] = acc;
    }
    __syncthreads();

    const float* xb = x3 + (size_t)b * 1024 * 64;   // (sp, c) channel-last

    float l[5][4];
    #pragma unroll
    for (int j = 0; j < 4; ++j) {
        const int pq = tid + j * 256;
        const float* xp = xb + (size_t)pq * 64;
        float a0 = beff[0], a1 = beff[1], a2 = beff[2], a3 = beff[3], a4 = beff[4];
        for (int c = 0; c < 64; ++c) {
            const float xv = (xp[c] - mus[c]) * rss[c];
            a0 += xv * weff[0 * 64 + c];
            a1 += xv * weff[1 * 64 + c];
            a2 += xv * weff[2 * 64 + c];
            a3 += xv * weff[3 * 64 + c];
            a4 += xv * weff[4 * 64 + c];
        }
        l[0][j] = a0; l[1][j] = a1; l[2][j] = a2; l[3][j] = a3; l[4][j] = a4;
    }

    for (int s = 0; s < 5; ++s) {
        float lm = -1e30f;
        #pragma unroll
        for (int j = 0; j < 4; ++j) lm = fmaxf(lm, l[s][j]);
        lm = block_reduce(lm, red, tid, true);

        float se = 0.0f, sex = 0.0f, sey = 0.0f;
        #pragma unroll
        for (int j = 0; j < 4; ++j) {
            const int pq  = tid + j * 256;
            const int row = pq >> 5, col = pq & 31;
            const float e = expf(l[s][j] - lm);
            se  += e;
            sex += e * (-1.0f + 2.0f * (float)col / 31.0f);
            sey += e * (-1.0f + 2.0f * (float)row / 31.0f);
        }
        se  = block_reduce(se,  red, tid, false);
        sex = block_reduce(sex, red, tid, false);
        sey = block_reduce(sey, red, tid, false);
        if (tid == 0) {
            out[((size_t)b * 5 + s) * 2 + 0] = sex / se;
            out[((size_t)b * 5 + s) * 2 + 1] = sey / se;
        }
    }
}

// ---------------------------------------------------------------------------
extern "C" void kernel_launch(void* const* d_in, const int* in_sizes, int n_in,
                              void* d_out, int out_size, void* d_ws, size_t ws_size,
                              hipStream_t stream) {
    const float* feat = (const float*)d_in[0];
    const int*   act  = (const int*)d_in[1];
    // d_in[2] = scene_idx: unused by the reference
    const float* fc_w = (const float*)d_in[3];
    const float* fc_b = (const float*)d_in[4];
    const float* gw0 = (const float*)d_in[5],  *gb0 = (const float*)d_in[6];
    const float* dw0 = (const float*)d_in[7],  *db0 = (const float*)d_in[8];
    const float* gw1 = (const float*)d_in[9],  *gb1 = (const float*)d_in[10];
    const float* dw1 = (const float*)d_in[11], *db1 = (const float*)d_in[12];
    const float* gw2 = (const float*)d_in[13], *gb2 = (const float*)d_in[14];
    const float* dw2 = (const float*)d_in[15], *db2 = (const float*)d_in[16];
    const float* hgw = (const float*)d_in[17], *hgb = (const float*)d_in[18];
    const float* hcw = (const float*)d_in[19], *hcb = (const float*)d_in[20];

    const int B = 512;
    float* ws = (float*)d_ws;
    // workspace layout (floats); activations are channel-last (b, spatial, C)
    float* x0  = ws;                           // B*16*512
    float* x1  = x0  + (size_t)B * 16 * 512;   // B*64*256
    float* x2  = x1  + (size_t)B * 64 * 256;   // B*256*128
    float* x3  = x2  + (size_t)B * 256 * 128;  // B*1024*64
    float* mu0 = x3  + (size_t)B * 1024 * 64;
    float* r0  = mu0 + (size_t)B * 512;
    float* mu1 = r0  + (size_t)B * 512;
    float* r1  = mu1 + (size_t)B * 256;
    float* mu2 = r1  + (size_t)B * 256;
    float* r2  = mu2 + (size_t)B * 128;
    float* mu3 = r2  + (size_t)B * 128;
    float* r3  = mu3 + (size_t)B * 64;
    float* ft  = r3  + (size_t)B * 64;         // B*16*256
    float* fwt = ft  + (size_t)B * 16 * 256;   // 512*256
    float* wp0 = fwt + (size_t)512 * 256;      // 9*256*512
    float* wp1 = wp0 + (size_t)9 * 256 * 512;  // 9*128*256
    float* wp2 = wp1 + (size_t)9 * 128 * 256;  // 9*64*128
    float* zbf = wp2 + (size_t)9 * 64 * 128;   // 512 zeros (padding row)

    // 0) layout packing + zero row (cheap, bandwidth-trivial)
    transpose_feat_kernel<<<(B * 16 * 256) / 256, 256, 0, stream>>>(feat, ft);
    transpose_fcw_kernel<<<(512 * 256) / 256, 256, 0, stream>>>(fc_w, fwt);
    pack_dw_kernel<<<(9 * 256 * 512) / 256, 256, 0, stream>>>(dw0, wp0, 256, 512);
    pack_dw_kernel<<<(9 * 128 * 256) / 256, 256, 0, stream>>>(dw1, wp1, 128, 256);
    pack_dw_kernel<<<(9 * 64 * 128) / 256, 256, 0, stream>>>(dw2, wp2, 64, 128);
    zero_fill_kernel<<<2, 256, 0, stream>>>(zbf, 512);

    // 1) fc + relu: (B,16,256) -> (B,16,512)
    fc_relu_kernel<<<dim3(8, B), 32, 0, stream>>>(ft, fwt, fc_b, x0);

    // 2) stage 0: (B,16,512) -> (B,64,256)
    gn_stats_kernel<<<(B * 512) / 256, 256, 0, stream>>>(x0, mu0, r0, 512, 16);
    gn_apply_kernel<<<(B * 16 * 512) / 256, 256, 0, stream>>>(x0, mu0, r0, gw0, gb0, 512, 16 * 512);
    deconv_stage_kernel<<<dim3(16, B / 2), 32, 0, stream>>>(x0, wp0, db0, zbf, x1, 512, 256, 4);

    // 3) stage 1: (B,64,256) -> (B,256,128)
    gn_stats_kernel<<<(B * 256) / 256, 256, 0, stream>>>(x1, mu1, r1, 256, 64);
    gn_apply_kernel<<<(B * 64 * 256) / 256, 256, 0, stream>>>(x1, mu1, r1, gw1, gb1, 256, 64 * 256);
    deconv_stage_kernel<<<dim3(32, B / 2), 32, 0, stream>>>(x1, wp1, db1, zbf, x2, 256, 128, 8);

    // 4) stage 2: (B,256,128) -> (B,1024,64)
    gn_stats_kernel<<<(B * 128) / 256, 256, 0, stream>>>(x2, mu2, r2, 128, 256);
    gn_apply_kernel<<<(B * 256 * 128) / 256, 256, 0, stream>>>(x2, mu2, r2, gw2, gb2, 128, 256 * 128);
    deconv_stage_kernel<<<dim3(64, B / 2), 32, 0, stream>>>(x2, wp2, db2, zbf, x3, 128, 64, 16);

    // 5) final norm stats + routed head + spatial softmax
    gn_stats_kernel<<<(B * 64) / 256, 256, 0, stream>>>(x3, mu3, r3, 64, 1024);
    head_softmax_kernel<<<B, 256, 0, stream>>>(x3, mu3, r3, act, hgw, hgb, hcw, hcb,
                                               (float*)d_out);
}